// Model_39676907887961
// MI455X (gfx1250) — compile-verified
//
#include <hip/hip_runtime.h>

typedef float v2f __attribute__((ext_vector_type(2)));
typedef float v4f __attribute__((ext_vector_type(4)));
typedef float v8f __attribute__((ext_vector_type(8)));
typedef int   vb128 __attribute__((vector_size(16)));   // pointee type for async b128

#define B_   2
#define H_   16
#define S_   2048
#define D_   128
#define KT   16          // key-tile size
#define NKT  (S_ / KT)   // 128 key tiles
#define WAVES 4
#define QW   16          // query rows per wave
#define TSTRIDE 132      // padded LDS row stride for 128-float rows (bank spread)
#define PSTRIDE 18       // padded LDS row stride for 16-float prob rows

#define GLOBAL_AS __attribute__((address_space(1)))
#define LDS_AS    __attribute__((address_space(3)))

#if defined(__gfx1250__) && __has_builtin(__builtin_amdgcn_global_load_async_to_lds_b128)
#define HAVE_ASYNC_LDS 1
#else
#define HAVE_ASYNC_LDS 0
#endif

__device__ __forceinline__ unsigned rotl32(unsigned x, int r) {
    return (x << r) | (x >> (32 - r));
}

// JAX threefry2x32 with key (0, 42) over flat index of the [B,H,S,S] tensor.
// N = 2^27 elements; counter pair is (j, j + 2^27), output word selected by
// idx >> 27. keep <=> uniform(bits) < 0.5 <=> sign bit of bits clear.
__device__ __forceinline__ bool keep_flag(unsigned idx) {
    const unsigned k0 = 0u, k1 = 42u;
    const unsigned k2 = 0x1BD11BDAu ^ k0 ^ k1;
    const unsigned half = 1u << 27;
    unsigned j   = idx & (half - 1u);
    unsigned sel = idx >> 27;
    unsigned x0 = j + k0;
    unsigned x1 = (j + half) + k1;
#define R4(a,b,c,d) \
    x0 += x1; x1 = rotl32(x1,a); x1 ^= x0; \
    x0 += x1; x1 = rotl32(x1,b); x1 ^= x0; \
    x0 += x1; x1 = rotl32(x1,c); x1 ^= x0; \
    x0 += x1; x1 = rotl32(x1,d); x1 ^= x0;
    R4(13,15,26,6)   x0 += k1; x1 += k2 + 1u;
    R4(17,29,16,24)  x0 += k2; x1 += k0 + 2u;
    R4(13,15,26,6)   x0 += k0; x1 += k1 + 3u;
    R4(17,29,16,24)  x0 += k1; x1 += k2 + 4u;
    R4(13,15,26,6)   x0 += k2; x1 += k0 + 5u;
#undef R4
    unsigned bits = sel ? x1 : x0;
    return (bits & 0x80000000u) == 0u;
}

__global__ __launch_bounds__(128) void fused_attn_kernel(
    const float* __restrict__ x1, const float* __restrict__ x2,
    const float* __restrict__ x3, float* __restrict__ out)
{
    __shared__ float x2t[KT * TSTRIDE];
    __shared__ float x1t[KT * TSTRIDE];
    __shared__ float pbuf[WAVES][KT * PSTRIDE];

    const int tid  = threadIdx.x;
    const int wv   = tid >> 5;
    const int lane = tid & 31;
    const int lmod = lane & 15;
    const int lhi  = lane >> 4;          // 0 or 1 (lane half)

    const int nqb = S_ / (WAVES * QW);   // 32 query blocks per (b,h)
    const int bh  = blockIdx.x / nqb;
    const int qb  = blockIdx.x % nqb;
    const int q0  = qb * (WAVES * QW) + wv * QW;

    const size_t basebh = (size_t)bh * S_ * D_;
    const float invx3 = 1.0f / x3[bh];

    // Resident A fragments: rows q0..q0+15 of x1, all 128 K values.
    // 32-bit A 16x4 layout: lane -> M = lane%16, K = (lane/16)*2 + c.
    v2f areg[32];
    {
        const float* ap = x1 + basebh + (size_t)(q0 + lmod) * D_ + lhi * 2;
#pragma unroll
        for (int ch = 0; ch < 32; ++ch)
            areg[ch] = *(const v2f*)(ap + ch * 4);
    }

    v8f acc[8];
#pragma unroll
    for (int i = 0; i < 8; ++i) acc[i] = {};
    float mrow[8], lrow[8];
#pragma unroll
    for (int r = 0; r < 8; ++r) { mrow[r] = -__builtin_inff(); lrow[r] = 0.0f; }

    const unsigned idx_bh = (unsigned)bh * (unsigned)(S_ * S_);

    for (int kt = 0; kt < NKT; ++kt) {
        const int k0 = kt * KT;
        __syncthreads();
        // Cooperative stage of the 16x128 key tiles of x2 and x1 (padded rows).
        {
            const float* s2 = x2 + basebh + (size_t)k0 * D_;
            const float* s1 = x1 + basebh + (size_t)k0 * D_;
#pragma unroll
            for (int i = 0; i < 4; ++i) {
                int e4  = tid + i * 128;   // 0..511 v4f chunks
                int row = e4 >> 5;
                int c4  = e4 & 31;
#if HAVE_ASYNC_LDS
                // CDNA5 async memory->LDS path (ASYNCcnt), no VGPR round trip.
                __builtin_amdgcn_global_load_async_to_lds_b128(
                    (GLOBAL_AS vb128*)(s2 + row * D_ + c4 * 4),
                    (LDS_AS vb128*)&x2t[row * TSTRIDE + c4 * 4], 0, 0);
                __builtin_amdgcn_global_load_async_to_lds_b128(
                    (GLOBAL_AS vb128*)(s1 + row * D_ + c4 * 4),
                    (LDS_AS vb128*)&x1t[row * TSTRIDE + c4 * 4], 0, 0);
#else
                v4f t2 = *(const v4f*)(s2 + row * D_ + c4 * 4);
                v4f t1 = *(const v4f*)(s1 + row * D_ + c4 * 4);
                *(v4f*)&x2t[row * TSTRIDE + c4 * 4] = t2;
                *(v4f*)&x1t[row * TSTRIDE + c4 * 4] = t1;
#endif
            }
        }
#if HAVE_ASYNC_LDS
#if __has_builtin(__builtin_amdgcn_s_wait_asynccnt)
        __builtin_amdgcn_s_wait_asynccnt(0);
#else
        asm volatile("s_wait_asynccnt 0" ::: "memory");
#endif
#endif
        __syncthreads();

        if (kt + 1 < NKT) {
            __builtin_prefetch(x2 + basebh + (size_t)(k0 + KT) * D_ + tid * 16, 0, 0);
            __builtin_prefetch(x1 + basebh + (size_t)(k0 + KT) * D_ + tid * 16, 0, 0);
        }

        // GEMM1: sc(16x16) = x1_q (16xD) * x2_k^T (Dx16), 32 K-chunks of 4.
        // B 4x16 layout: lane -> N = lane%16, K = (lane/16)*2 + c.
        v8f sc = {};
#pragma unroll
        for (int ch = 0; ch < 32; ++ch) {
            v2f b = *(const v2f*)&x2t[lmod * TSTRIDE + ch * 4 + lhi * 2];
            sc = __builtin_amdgcn_wmma_f32_16x16x4_f32(
                false, areg[ch], false, b, (short)0, sc, false, false);
        }

        // Online softmax + dropout; C/D layout: comp r -> row (r + lhi*8), col lmod.
#pragma unroll
        for (int r = 0; r < 8; ++r) {
            float s = sc[r] * invx3;
            float tm = s;
            tm = fmaxf(tm, __shfl_xor(tm, 1, 32));
            tm = fmaxf(tm, __shfl_xor(tm, 2, 32));
            tm = fmaxf(tm, __shfl_xor(tm, 4, 32));
            tm = fmaxf(tm, __shfl_xor(tm, 8, 32));
            float mnew  = fmaxf(mrow[r], tm);
            float alpha = __expf(mrow[r] - mnew);
            float p     = __expf(s - mnew);
            float ts = p;
            ts += __shfl_xor(ts, 1, 32);
            ts += __shfl_xor(ts, 2, 32);
            ts += __shfl_xor(ts, 4, 32);
            ts += __shfl_xor(ts, 8, 32);
            lrow[r] = lrow[r] * alpha + ts;   // denominator: pre-dropout probs
            mrow[r] = mnew;
#pragma unroll
            for (int dt = 0; dt < 8; ++dt) acc[dt][r] *= alpha;
            unsigned qg  = (unsigned)(q0 + r + lhi * 8);
            unsigned kg  = (unsigned)(k0 + lmod);
            unsigned idx = idx_bh + qg * (unsigned)S_ + kg;
            float pm = keep_flag(idx) ? 2.0f * p : 0.0f;   // /(1-p) rescale
            pbuf[wv][(r + lhi * 8) * PSTRIDE + lmod] = pm;
        }
        asm volatile("s_wait_dscnt 0" ::: "memory");   // wave-private LDS RAW

        // GEMM2: acc(16x128) += probs(16x16) * x1_k (16x128).
#pragma unroll
        for (int dt = 0; dt < 8; ++dt) {
#pragma unroll
            for (int kk = 0; kk < 4; ++kk) {
                v2f a = *(const v2f*)&pbuf[wv][lmod * PSTRIDE + kk * 4 + lhi * 2];
                int K = kk * 4 + lhi * 2;
                v2f b = { x1t[(K + 0) * TSTRIDE + dt * 16 + lmod],
                          x1t[(K + 1) * TSTRIDE + dt * 16 + lmod] };
                acc[dt] = __builtin_amdgcn_wmma_f32_16x16x4_f32(
                    false, a, false, b, (short)0, acc[dt], false, false);
            }
        }
    }

    // Epilogue: out = acc / l + x2 residual.
#pragma unroll
    for (int r = 0; r < 8; ++r) {
        float invl = 1.0f / lrow[r];
        int q = q0 + r + lhi * 8;
        const float* x2row = x2 + basebh + (size_t)q * D_;
        float* orow        = out + basebh + (size_t)q * D_;
#pragma unroll
        for (int dt = 0; dt < 8; ++dt) {
            int d = dt * 16 + lmod;
            orow[d] = acc[dt][r] * invl + x2row[d];
        }
    }
}

extern "C" void kernel_launch(void* const* d_in, const int* in_sizes, int n_in,
                              void* d_out, int out_size, void* d_ws, size_t ws_size,
                              hipStream_t stream) {
    const float* x1 = (const float*)d_in[0];
    const float* x2 = (const float*)d_in[1];
    const float* x3 = (const float*)d_in[2];
    float* out = (float*)d_out;
    (void)in_sizes; (void)n_in; (void)out_size; (void)d_ws; (void)ws_size;
    dim3 grid(B_ * H_ * (S_ / (WAVES * QW)));   // 2*16*32 = 1024 workgroups
    fused_attn_kernel<<<grid, WAVES * 32, 0, stream>>>(x1, x2, x3, out);
}